// TritonRnntLoss_9723805958619
// MI455X (gfx1250) — compile-verified
//
#include <hip/hip_runtime.h>
#include <hip/hip_bf16.h>
#include <math.h>

// Problem constants (match reference)
#define B_ 8
#define T_ 160
#define U_ 80
#define V_ 512
#define NROWS (B_ * T_ * (U_ + 1))      // 103680 softmax rows
#define GRID1 1620                      // 1620 blocks * 8 waves = 12960 waves -> 8 rows/wave
#define FASTEMIT_LAMBDA 0.0f

// ---------------------------------------------------------------------------
// CDNA5 async global->LDS copy (tracked by ASYNCcnt).
// Builtin signature (from hipcc diagnostic): (v4i AS1*, v4i AS3*, imm, cpol).
// ---------------------------------------------------------------------------
typedef int v4i __attribute__((ext_vector_type(4)));
typedef v4i __attribute__((address_space(1))) gv4i;
typedef v4i __attribute__((address_space(3))) lv4i;

__device__ __forceinline__ void async_copy_b128(const float* g, float* l) {
#if __has_builtin(__builtin_amdgcn_global_load_async_to_lds_b128)
  __builtin_amdgcn_global_load_async_to_lds_b128(
      (gv4i*)g, (lv4i*)l, /*imm offset*/ 0, /*cpol*/ 0);
#else
  *reinterpret_cast<float4*>(l) = *reinterpret_cast<const float4*>(g);
#endif
}

template <int N>
__device__ __forceinline__ void s_wait_asynccnt() {
#if __has_builtin(__builtin_amdgcn_s_wait_asynccnt)
  __builtin_amdgcn_s_wait_asynccnt(N);
#else
  asm volatile("s_wait_asynccnt %0" ::"i"(N) : "memory");
#endif
}

// ---------------------------------------------------------------------------
// Kernel 1: per-(b,t,u) log-softmax statistics.
// One wave32 per 512-float row. Double-buffered async loads into LDS:
//   - 4 x global_load_async_to_lds_b128 per row (32 lanes * 16B = 512B each)
//   - wave reduction (max, sum-exp) via __shfl_xor
//   - lane 0 writes blank_lp[r] and label_lp (gather is a direct LDS read)
// ---------------------------------------------------------------------------
__global__ void __launch_bounds__(256) rnnt_logprobs_kernel(
    const float* __restrict__ acts, const int* __restrict__ labels,
    float* __restrict__ blank_lp, float* __restrict__ label_lp) {
  __shared__ float lds[2 * 8 * V_];  // 8 waves * 2 buffers * 2KB = 32 KB

  const int lane = threadIdx.x & 31;
  const int wave = threadIdx.x >> 5;
  const int gw = blockIdx.x * 8 + wave;
  const int nw = gridDim.x * 8;

  float* const buf0 = &lds[(wave * 2 + 0) * V_];
  float* const buf1 = &lds[(wave * 2 + 1) * V_];

  auto prefetch = [&](int row, float* buf) {
    const float* src = acts + (size_t)row * V_;
#pragma unroll
    for (int k = 0; k < 4; ++k) {
      const int off = k * 128 + lane * 4;  // floats
      async_copy_b128(src + off, buf + off);
    }
  };

  int r = gw;
  if (r < NROWS) prefetch(r, buf0);
  int parity = 0;

  for (; r < NROWS; r += nw) {
    const int rn = r + nw;
    if (rn < NROWS) {
      prefetch(rn, parity ? buf0 : buf1);  // fill the other buffer
      s_wait_asynccnt<4>();                // current row's 4 copies done
    } else {
      s_wait_asynccnt<0>();
    }
    float* const buf = parity ? buf1 : buf0;

    // Each lane owns 16 elements (4 x float4 from LDS).
    float v[16];
    float vmax = -INFINITY;
#pragma unroll
    for (int k = 0; k < 4; ++k) {
      const float4 q = *reinterpret_cast<const float4*>(buf + k * 128 + lane * 4);
      v[k * 4 + 0] = q.x; v[k * 4 + 1] = q.y;
      v[k * 4 + 2] = q.z; v[k * 4 + 3] = q.w;
      vmax = fmaxf(vmax, fmaxf(fmaxf(q.x, q.y), fmaxf(q.z, q.w)));
    }
#pragma unroll
    for (int off = 16; off > 0; off >>= 1)
      vmax = fmaxf(vmax, __shfl_xor(vmax, off, 32));

    float s = 0.0f;
#pragma unroll
    for (int i = 0; i < 16; ++i) s += expf(v[i] - vmax);
#pragma unroll
    for (int off = 16; off > 0; off >>= 1)
      s += __shfl_xor(s, off, 32);

    const float lse = vmax + logf(s);

    if (lane == 0) {
      const int b = r / (T_ * (U_ + 1));
      const int rem = r - b * (T_ * (U_ + 1));
      const int t = rem / (U_ + 1);
      const int u = rem - t * (U_ + 1);
      blank_lp[r] = buf[0] - lse;  // BLANK == 0
      if (u < U_) {
        const int lab = labels[b * U_ + u];       // in [1, V)
        label_lp[(b * T_ + t) * U_ + u] = buf[lab] - lse;
      }
    }
    parity ^= 1;
  }
}

// ---------------------------------------------------------------------------
// Kernel 2: forward DP (alpha) per batch element, anti-diagonal wavefront.
// Full alpha lattice lives in LDS (160*81*4 = 51.8 KB << 320 KB/WGP).
// ---------------------------------------------------------------------------
__device__ __forceinline__ float logaddexpf_(float a, float b) {
  const float mx = fmaxf(a, b);
  const float mn = fminf(a, b);
  return mx + log1pf(expf(mn - mx));
}

__global__ void __launch_bounds__(96) rnnt_alpha_kernel(
    const float* __restrict__ blank_lp, const float* __restrict__ label_lp,
    const int* __restrict__ act_lens, const int* __restrict__ label_lens,
    float* __restrict__ out) {
  __shared__ float alpha[T_ * (U_ + 1)];  // 51840 bytes

  const int b = blockIdx.x;
  const float* const blp = blank_lp + (size_t)b * T_ * (U_ + 1);
  const float* const llp = label_lp + (size_t)b * T_ * U_;
  const int u = threadIdx.x;

  for (int d = 0; d < T_ + U_; ++d) {  // d = t + u, max = (T-1)+U = 239
    const int t = d - u;
    if (u <= U_ && t >= 0 && t < T_) {
      float a;
      if (t == 0 && u == 0) {
        a = 0.0f;
      } else if (t == 0) {
        a = alpha[u - 1] + llp[u - 1];                       // cumsum row 0
      } else if (u == 0) {
        a = alpha[(t - 1) * (U_ + 1)] + blp[(t - 1) * (U_ + 1)];
      } else {
        const float x = alpha[(t - 1) * (U_ + 1) + u] + blp[(t - 1) * (U_ + 1) + u];
        const float y = alpha[t * (U_ + 1) + (u - 1)] + llp[t * U_ + (u - 1)];
        a = logaddexpf_(x, y);
      }
      alpha[t * (U_ + 1) + u] = a;
    }
    __syncthreads();
  }

  if (threadIdx.x == 0) {
    const int ti = act_lens[b] - 1;
    const int lu = label_lens[b];
    const float ll = alpha[ti * (U_ + 1) + lu] + blp[ti * (U_ + 1) + lu];
    out[b] = -(1.0f + FASTEMIT_LAMBDA) * ll;
  }
}

// ---------------------------------------------------------------------------
// Launch: inputs are (acts f32, labels i32, act_lens i32, label_lens i32);
// output is 8 f32 losses. Workspace holds blank_lp then label_lp.
// ---------------------------------------------------------------------------
extern "C" void kernel_launch(void* const* d_in, const int* in_sizes, int n_in,
                              void* d_out, int out_size, void* d_ws, size_t ws_size,
                              hipStream_t stream) {
  const float* acts = (const float*)d_in[0];
  const int* labels = (const int*)d_in[1];
  const int* act_lens = (const int*)d_in[2];
  const int* label_lens = (const int*)d_in[3];

  float* blank_lp = (float*)d_ws;                 // NROWS floats
  float* label_lp = blank_lp + NROWS;             // B*T*U floats
  float* out = (float*)d_out;                     // B floats

  rnnt_logprobs_kernel<<<GRID1, 256, 0, stream>>>(acts, labels, blank_lp, label_lp);
  rnnt_alpha_kernel<<<B_, 96, 0, stream>>>(blank_lp, label_lp, act_lens, label_lens, out);
}